// FlaxStaticEmbeddingModule_3977139716495
// MI455X (gfx1250) — compile-verified
//
#include <hip/hip_runtime.h>

// one_hot(arange(N), N) @ W  ==  W, computed per-16x16-tile as I16 x W_tile
// on the CDNA5 WMMA f32 pipe: D = sum_k A_k(16x4) * B_k(4x16), where
//   A_k = W_tile[:, 4k:4k+4]   (loaded from global, b64 per lane)
//   B_k = identity slice: B_k[r][n] = (n == 4k + r)   (built in registers)
// Memory-bound: 16 MiB total traffic -> ~0.7us at 23.3 TB/s; WMMA cost ~67 MFLOP
// is negligible, so the matrix-engine path is free.

typedef __attribute__((ext_vector_type(2))) float v2f;
typedef __attribute__((ext_vector_type(8))) float v8f;

#define NUM_EMB 16384
#define EMB_DIM 128

__global__ __launch_bounds__(256)
void onehot_matmul_wmma_f32(const float* __restrict__ W, float* __restrict__ out) {
    const int lane = threadIdx.x & 31;
    const int wave = threadIdx.x >> 5;     // 0..7 -> which 16-col tile
    const int tm   = blockIdx.x * 16;      // tile row base (1024 blocks)
    const int tn   = wave * 16;            // tile col base

    const int half = lane >> 4;            // 0: lanes 0-15, 1: lanes 16-31
    const int l15  = lane & 15;

    // ---- B_k = 4x16 identity slices, synthesized once, reused for the tile ----
    // 32-bit B layout (mirrors C/D row striping): VGPR v, lane L:
    //   K = v + 2*(L>=16), N = L&15
    v2f bmat[4];
    #pragma unroll
    for (int k = 0; k < 4; ++k) {
        #pragma unroll
        for (int v = 0; v < 2; ++v) {
            const int r = v + 2 * half;               // K index this lane/VGPR holds
            bmat[k][v] = (l15 == 4 * k + r) ? 1.0f : 0.0f;
        }
    }

    // ---- A_k = W[tm:tm+16, tn+4k : tn+4k+4], 16x4 f32 ----
    // 32-bit A layout: lane L: M = L&15; VGPR0 holds K=0(lo)/K=2(hi),
    // VGPR1 holds K=1(lo)/K=3(hi) -> each lane's pair is contiguous in memory.
    const int row = tm + l15;
    const float* base = W + (size_t)row * EMB_DIM + tn + 2 * half;
    v2f amat[4];
    #pragma unroll
    for (int k = 0; k < 4; ++k) {
        amat[k] = *(const v2f*)(base + 4 * k);        // global_load_b64
    }

    // ---- D = I16 * W_tile via 4 chained f32 WMMAs (K=16 total) ----
    v8f acc = {};
    #pragma unroll
    for (int k = 0; k < 4; ++k) {
        acc = __builtin_amdgcn_wmma_f32_16x16x4_f32(
            /*neg_a=*/false, amat[k],
            /*neg_b=*/false, bmat[k],
            /*c_mod=*/(short)0, acc,
            /*reuse_a=*/false, /*reuse_b=*/false);
    }

    // ---- store D: VGPR v, lane L: M = v + 8*half, N = l15 ----
    #pragma unroll
    for (int v = 0; v < 8; ++v) {
        const int m = v + 8 * half;
        out[(size_t)(tm + m) * EMB_DIM + tn + l15] = acc[v];
    }
}

extern "C" void kernel_launch(void* const* d_in, const int* in_sizes, int n_in,
                              void* d_out, int out_size, void* d_ws, size_t ws_size,
                              hipStream_t stream) {
    const float* W = (const float*)d_in[0];
    float* out = (float*)d_out;
    (void)in_sizes; (void)n_in; (void)out_size; (void)d_ws; (void)ws_size;

    dim3 grid(NUM_EMB / 16);   // 1024 blocks, each = one 16-row strip
    dim3 block(256);           // 8 waves -> 8 column tiles (full 128-wide row)
    onehot_matmul_wmma_f32<<<grid, block, 0, stream>>>(W, out);
}